// Curating_of_attention_mask_8813272891856
// MI455X (gfx1250) — compile-verified
//
#include <hip/hip_runtime.h>
#include <hip/hip_bf16.h>

// Problem constants (reference is fixed-shape: (3, 1280, 1280), PATCH=16)
#define CCH 3
#define HH  1280
#define WW  1280
#define PS  16
#define HP  (HH / PS)        // 80
#define WP  (WW / PS)        // 80
#define NPATCH (HP * WP)     // 6400
#define NTILE  (NPATCH / 16) // 400
#define GF_STRIDE 12         // 9 gram entries padded to 12 (3 x K=4 WMMA chunks)

typedef float v2f __attribute__((ext_vector_type(2)));
typedef float v8f __attribute__((ext_vector_type(8)));

// ---- monotonic float <-> uint mapping so uint atomicMin/Max order like floats ----
__device__ __forceinline__ unsigned int fkey(float f) {
    unsigned int u = __float_as_uint(f);
    return (u & 0x80000000u) ? ~u : (u | 0x80000000u);
}
__device__ __forceinline__ float funkey(unsigned int k) {
    unsigned int u = (k & 0x80000000u) ? (k ^ 0x80000000u) : ~k;
    return __uint_as_float(u);
}

// ---------------------------------------------------------------------------
// Kernel 0: reset the min/max cells (ws is not re-poisoned between replays)
// ---------------------------------------------------------------------------
__global__ void k_init(unsigned int* __restrict__ mm) {
    mm[0] = 0xFFFFFFFFu;  // min accumulator: largest key
    mm[1] = 0x00000000u;  // max accumulator: smallest key
}

// ---------------------------------------------------------------------------
// Kernel 1: per-patch 3x3 Gram. One wave32 per patch, 8 patches per block.
//   gf[p*12 + 0..8] = flattened symmetric G / 768, [9..11] = 0 (K padding)
//   sq[p]           = sum(Gf[p]^2)
// ---------------------------------------------------------------------------
__global__ void k_gram(const float* __restrict__ img,
                       float* __restrict__ gf,
                       float* __restrict__ sq) {
    const int wave = threadIdx.x >> 5;
    const int lane = threadIdx.x & 31;
    const int p = blockIdx.x * 8 + wave;
    if (p >= NPATCH) return;

    const int py = p / WP, px = p % WP;
    const float* base = img + (size_t)py * PS * WW + (size_t)px * PS;

    const int col  = lane & 15;   // column within patch
    const int row0 = lane >> 4;   // 0 or 1; rows row0 + 2*i, i=0..7

    float s00 = 0.f, s01 = 0.f, s02 = 0.f, s11 = 0.f, s12 = 0.f, s22 = 0.f;
#pragma unroll
    for (int i = 0; i < 8; ++i) {
        const int r = row0 + 2 * i;
        const float* pix = base + (size_t)r * WW + col;
        const float a = pix[0];
        const float b = pix[(size_t)HH * WW];
        const float c = pix[2 * (size_t)HH * WW];
        s00 += a * a; s01 += a * b; s02 += a * c;
        s11 += b * b; s12 += b * c; s22 += c * c;
    }
    // wave32 tree reduction
#pragma unroll
    for (int off = 16; off > 0; off >>= 1) {
        s00 += __shfl_down(s00, off, 32);
        s01 += __shfl_down(s01, off, 32);
        s02 += __shfl_down(s02, off, 32);
        s11 += __shfl_down(s11, off, 32);
        s12 += __shfl_down(s12, off, 32);
        s22 += __shfl_down(s22, off, 32);
    }
    if (lane == 0) {
        const float inv = 1.0f / (float)(CCH * PS * PS);  // 1/768
        const float g00 = s00 * inv, g01 = s01 * inv, g02 = s02 * inv;
        const float g11 = s11 * inv, g12 = s12 * inv, g22 = s22 * inv;
        float* row = gf + (size_t)p * GF_STRIDE;
        row[0] = g00; row[1] = g01; row[2] = g02;
        row[3] = g01; row[4] = g11; row[5] = g12;
        row[6] = g02; row[7] = g12; row[8] = g22;
        row[9] = 0.f; row[10] = 0.f; row[11] = 0.f;
        sq[p] = g00 * g00 + 2.f * g01 * g01 + 2.f * g02 * g02 +
                g11 * g11 + 2.f * g12 * g12 + g22 * g22;
    }
}

// ---------------------------------------------------------------------------
// 16x16 tile of Gf . Gf^T via V_WMMA_F32_16X16X4_F32, K padded 9 -> 12.
// A frag (16x4 f32): lane holds M=lane&15; VGPR v holds K = k0 + v + 2*(lane>>4).
// B frag (4x16 f32): mirrors A with N=lane&15 -> identical index math since
// both operands come from the same (symmetric-product) Gf table.
// ---------------------------------------------------------------------------
__device__ __forceinline__ v8f tile_inner(const float* __restrict__ gf,
                                          int g0, int n0, int lane) {
    const int m  = lane & 15;
    const int hi = (lane >> 4) << 1;  // 0 or 2
    const float* arow = gf + (size_t)(g0 + m) * GF_STRIDE + hi;
    const float* brow = gf + (size_t)(n0 + m) * GF_STRIDE + hi;
    v8f acc = {};
#pragma unroll
    for (int k0 = 0; k0 < GF_STRIDE; k0 += 4) {
        v2f a, b;
        a.x = arow[k0]; a.y = arow[k0 + 1];
        b.x = brow[k0]; b.y = brow[k0 + 1];
        acc = __builtin_amdgcn_wmma_f32_16x16x4_f32(
            /*neg_a=*/false, a, /*neg_b=*/false, b,
            /*c_mod=*/(short)0, acc, /*reuse_a=*/false, /*reuse_b=*/false);
    }
    return acc;
}

// ---------------------------------------------------------------------------
// Kernel 2: compute dist tiles, reduce global min/max. No dist writes.
// dist is symmetric -> only tiles with tg <= tn are computed (upper triangle
// incl. diagonal covers every distinct value); other waves contribute +/-inf.
// One wave per 16x16 tile; 8 waves (tiles) per block.
// ---------------------------------------------------------------------------
__global__ void k_minmax(const float* __restrict__ gf,
                         const float* __restrict__ sq,
                         unsigned int* __restrict__ mm) {
    __shared__ float smin[8], smax[8];
    const int wave = threadIdx.x >> 5;
    const int lane = threadIdx.x & 31;
    const int tile = blockIdx.x * 8 + wave;

    float lmin = __builtin_huge_valf();
    float lmax = -__builtin_huge_valf();
    if (tile < NTILE * NTILE) {
        const int tg = tile / NTILE, tn = tile % NTILE;
        if (tg <= tn) {  // symmetry: upper triangle is sufficient for min/max
            const int g0 = tg * 16, n0 = tn * 16;
            const v8f c = tile_inner(gf, g0, n0, lane);
            const int hi8 = (lane >> 4) << 3;  // 0 or 8
            const float sn = sq[n0 + (lane & 15)];
#pragma unroll
            for (int r = 0; r < 8; ++r) {
                const float d = (sq[g0 + r + hi8] + sn - 2.0f * c[r]) * (1.0f / 9.0f);
                lmin = fminf(lmin, d);
                lmax = fmaxf(lmax, d);
            }
        }
    }
#pragma unroll
    for (int off = 16; off > 0; off >>= 1) {
        lmin = fminf(lmin, __shfl_down(lmin, off, 32));
        lmax = fmaxf(lmax, __shfl_down(lmax, off, 32));
    }
    if (lane == 0) { smin[wave] = lmin; smax[wave] = lmax; }
    __syncthreads();
    if (threadIdx.x == 0) {
        float m0 = smin[0], m1 = smax[0];
#pragma unroll
        for (int i = 1; i < 8; ++i) {
            m0 = fminf(m0, smin[i]);
            m1 = fmaxf(m1, smax[i]);
        }
        atomicMin(&mm[0], fkey(m0));
        atomicMax(&mm[1], fkey(m1));
    }
}

// ---------------------------------------------------------------------------
// Kernel 3: recompute dist tiles, normalize, write the 164 MB output once.
// Output is write-once/never-re-read -> non-temporal stores (TH=NT) so the
// streaming write doesn't churn L2; the hot 307 KB Gf table stays cached.
// ---------------------------------------------------------------------------
__global__ void k_out(const float* __restrict__ gf,
                      const float* __restrict__ sq,
                      const unsigned int* __restrict__ mm,
                      float* __restrict__ out) {
    const int wave = threadIdx.x >> 5;
    const int lane = threadIdx.x & 31;
    const int tile = blockIdx.x * 8 + wave;
    if (tile >= NTILE * NTILE) return;

    const float mn = funkey(mm[0]);
    const float mx = funkey(mm[1]);
    const float inv = 1.0f / (mx - mn);

    const int tg = tile / NTILE, tn = tile % NTILE;
    const int g0 = tg * 16, n0 = tn * 16;
    const v8f c = tile_inner(gf, g0, n0, lane);
    const int hi8 = (lane >> 4) << 3;
    const int col = n0 + (lane & 15);
    const float sn = sq[col];
#pragma unroll
    for (int r = 0; r < 8; ++r) {
        const int row = g0 + r + hi8;
        const float d = (sq[row] + sn - 2.0f * c[r]) * (1.0f / 9.0f);
        __builtin_nontemporal_store((d - mn) * inv,
                                    out + (size_t)row * NPATCH + col);
    }
}

// ---------------------------------------------------------------------------
extern "C" void kernel_launch(void* const* d_in, const int* in_sizes, int n_in,
                              void* d_out, int out_size, void* d_ws, size_t ws_size,
                              hipStream_t stream) {
    (void)in_sizes; (void)n_in; (void)out_size;

    const float* img = (const float*)d_in[0];
    float* out = (float*)d_out;

    // workspace layout: gf[N*12] | sq[N] | mm[2] (uint)
    const size_t gf_elems = (size_t)NPATCH * GF_STRIDE;
    const size_t needed = (gf_elems + NPATCH) * sizeof(float) + 2 * sizeof(unsigned int);
    if (ws_size < needed) return;

    float* gf = (float*)d_ws;
    float* sq = gf + gf_elems;
    unsigned int* mm = (unsigned int*)(sq + NPATCH);

    k_init<<<1, 1, 0, stream>>>(mm);

    // 6400 patches / 8 waves-per-block = 800 blocks
    k_gram<<<NPATCH / 8, 256, 0, stream>>>(img, gf, sq);

    // 400*400 = 160000 tiles / 8 per block = 20000 blocks
    const int ntiles = NTILE * NTILE;
    k_minmax<<<(ntiles + 7) / 8, 256, 0, stream>>>(gf, sq, mm);
    k_out<<<(ntiles + 7) / 8, 256, 0, stream>>>(gf, sq, mm, out);
}